// GCNwithMlp_12360915878362
// MI455X (gfx1250) — compile-verified
//
#include <hip/hip_runtime.h>
#include <math.h>

typedef __attribute__((ext_vector_type(2))) float v2f;
typedef __attribute__((ext_vector_type(8))) float v8f;

#define DFEAT 128

// ---------------------------------------------------------------------------
// Degree / normalization kernels (PyG GCN norm with self loops)
// ---------------------------------------------------------------------------
__global__ void deg_init_kernel(float* __restrict__ deg, int n) {
    int i = blockIdx.x * blockDim.x + threadIdx.x;
    if (i < n) deg[i] = 1.0f;  // self loop contributes 1 to every node's degree
}

__global__ void deg_count_kernel(const long long* __restrict__ dst,
                                 float* __restrict__ deg, int e) {
    int i = blockIdx.x * blockDim.x + threadIdx.x;
    if (i < e) {
        int d = (int)dst[i];
        __hip_atomic_fetch_add(&deg[d], 1.0f, __ATOMIC_RELAXED, __HIP_MEMORY_SCOPE_AGENT);
    }
}

__global__ void deg_finalize_kernel(float* __restrict__ deg, int n) {
    int i = blockIdx.x * blockDim.x + threadIdx.x;
    if (i < n) deg[i] = rsqrtf(deg[i]);  // deg >= 1 always
}

// ---------------------------------------------------------------------------
// Tiled fp32 WMMA GEMM:  C[nrows x COLS] = A[nrows x 128] * W[128 x COLS]
// One block = 16*RT rows x COLS cols. One wave per 16-col tile, RT row tiles
// per wave so the B fragment is loaded once per k-step and reused RT times.
// LDS: A strip 16*RT*128*4 B + W 128*COLS*4 B (96 KB for COLS=128, RT=4).
// Optional fused epilogue (C2 != null): C2 = dinv[r]^2 * val + bc2[c]
// (self-loop init of the GCN aggregation buffer).
// ---------------------------------------------------------------------------
template <int COLS, int RT>
__global__ __launch_bounds__(32 * (COLS / 16))
void gemm_wmma_kernel(const float* __restrict__ A, const float* __restrict__ W,
                      const float* __restrict__ bias, float* __restrict__ C,
                      int nrows, int do_relu,
                      const float* __restrict__ dinv,
                      const float* __restrict__ bc2,
                      float* __restrict__ C2) {
    constexpr int ROWS = 16 * RT;
    __shared__ float sA[ROWS * DFEAT];
    __shared__ float sW[DFEAT * COLS];

    const int row_base = blockIdx.x * ROWS;

    // cooperative LDS fill (b128 loads)
    for (int idx = threadIdx.x * 4; idx < ROWS * DFEAT; idx += blockDim.x * 4) {
        int r = row_base + (idx >> 7);
        float4 v;
        if (r < nrows) v = *(const float4*)&A[(size_t)r * DFEAT + (idx & 127)];
        else           v = make_float4(0.f, 0.f, 0.f, 0.f);
        *(float4*)&sA[idx] = v;
    }
    for (int idx = threadIdx.x * 4; idx < DFEAT * COLS; idx += blockDim.x * 4) {
        *(float4*)&sW[idx] = *(const float4*)&W[idx];
    }
    __syncthreads();

    const int wave     = threadIdx.x >> 5;     // one 16-col tile per wave
    const int lane     = threadIdx.x & 31;
    const int half     = lane >> 4;            // 0: lanes 0-15, 1: lanes 16-31
    const int l15      = lane & 15;
    const int col_base = wave * 16;

    v8f acc[RT];
    #pragma unroll
    for (int rt = 0; rt < RT; ++rt) acc[rt] = (v8f){};

    #pragma unroll
    for (int k = 0; k < DFEAT; k += 4) {
        const int ka = k + 2 * half;
        // B fragment (4x16 f32): shared across all RT row tiles
        v2f b;
        b.x = sW[ka * COLS + col_base + l15];
        b.y = sW[(ka + 1) * COLS + col_base + l15];
        #pragma unroll
        for (int rt = 0; rt < RT; ++rt) {
            // A fragment (16x4 f32): lanes 0-15 K={k,k+1}, lanes 16-31 K={k+2,k+3}
            v2f a;
            a.x = sA[(rt * 16 + l15) * DFEAT + ka];
            a.y = sA[(rt * 16 + l15) * DFEAT + ka + 1];
            acc[rt] = __builtin_amdgcn_wmma_f32_16x16x4_f32(
                false, a, false, b, (short)0, acc[rt], false, false);
        }
    }

    // epilogue: C layout — VGPR i: lanes 0-15 -> M=i, lanes 16-31 -> M=i+8
    const int cidx = col_base + l15;
    const float bv  = bias ? bias[cidx] : 0.0f;
    const float bv2 = C2 ? bc2[cidx] : 0.0f;
    #pragma unroll
    for (int rt = 0; rt < RT; ++rt) {
        #pragma unroll
        for (int i = 0; i < 8; ++i) {
            const int r = row_base + rt * 16 + i + 8 * half;
            if (r < nrows) {
                float v = acc[rt][i] + bv;
                if (do_relu) v = fmaxf(v, 0.0f);
                C[(size_t)r * COLS + cidx] = v;
                if (C2) {
                    float s = dinv[r];
                    C2[(size_t)r * COLS + cidx] = fmaf(v, s * s, bv2);
                }
            }
        }
    }
}

// ---------------------------------------------------------------------------
// Per-edge scatter: out[dst] += dinv[src]*dinv[dst] * h[src]
// One wave per edge; lane handles 4 features (b128 load + 4 f32 atomics).
// Everything is L2-resident (h = 20.5 MB << 192 MB L2).
// ---------------------------------------------------------------------------
__global__ void agg_edges_kernel(const float* __restrict__ h,
                                 const float* __restrict__ dinv,
                                 const long long* __restrict__ src,
                                 const long long* __restrict__ dst,
                                 float* __restrict__ out, int e) {
    long long t = (long long)blockIdx.x * blockDim.x + threadIdx.x;  // e*32 threads
    long long ei = t >> 5;
    if (ei >= e) return;
    int f = (int)(t & 31) * 4;
    int s = (int)src[ei];
    int d = (int)dst[ei];
    float w = dinv[s] * dinv[d];
    float4 v = *(const float4*)&h[(size_t)s * DFEAT + f];
    float* o = out + (size_t)d * DFEAT + f;
    __hip_atomic_fetch_add(o + 0, v.x * w, __ATOMIC_RELAXED, __HIP_MEMORY_SCOPE_AGENT);
    __hip_atomic_fetch_add(o + 1, v.y * w, __ATOMIC_RELAXED, __HIP_MEMORY_SCOPE_AGENT);
    __hip_atomic_fetch_add(o + 2, v.z * w, __ATOMIC_RELAXED, __HIP_MEMORY_SCOPE_AGENT);
    __hip_atomic_fetch_add(o + 3, v.w * w, __ATOMIC_RELAXED, __HIP_MEMORY_SCOPE_AGENT);
}

// ---------------------------------------------------------------------------
// Head: out[i] = sigmoid(dot(h64[i,:], Wf2) + bf2)
// ---------------------------------------------------------------------------
__global__ void head_kernel(const float* __restrict__ h64,
                            const float* __restrict__ Wf2,
                            const float* __restrict__ bf2,
                            float* __restrict__ out, int n) {
    int i = blockIdx.x * blockDim.x + threadIdx.x;
    if (i >= n) return;
    float acc = bf2[0];
    #pragma unroll
    for (int k = 0; k < 64; k += 4) {
        float4 v = *(const float4*)&h64[(size_t)i * 64 + k];
        float4 w = *(const float4*)&Wf2[k];
        acc = fmaf(v.x, w.x, acc);
        acc = fmaf(v.y, w.y, acc);
        acc = fmaf(v.z, w.z, acc);
        acc = fmaf(v.w, w.w, acc);
    }
    out[i] = 1.0f / (1.0f + __expf(-acc));
}

// ---------------------------------------------------------------------------
// Launch
// ---------------------------------------------------------------------------
extern "C" void kernel_launch(void* const* d_in, const int* in_sizes, int n_in,
                              void* d_out, int out_size, void* d_ws, size_t ws_size,
                              hipStream_t stream) {
    (void)n_in; (void)out_size; (void)ws_size;

    const float* x  = (const float*)d_in[0];
    const long long* ei = (const long long*)d_in[1];
    const int N = in_sizes[0] / DFEAT;
    const int E = in_sizes[1] / 2;
    const long long* esrc = ei;
    const long long* edst = ei + E;

    const float* Wc[3]; const float* bc[3];
    const float* Wa[3]; const float* ba[3];
    const float* Wb[3]; const float* bb[3];
    for (int l = 0; l < 3; ++l) {
        Wc[l] = (const float*)d_in[2 + 6 * l + 0];
        bc[l] = (const float*)d_in[2 + 6 * l + 1];
        Wa[l] = (const float*)d_in[2 + 6 * l + 2];
        ba[l] = (const float*)d_in[2 + 6 * l + 3];
        Wb[l] = (const float*)d_in[2 + 6 * l + 4];
        bb[l] = (const float*)d_in[2 + 6 * l + 5];
    }
    const float* Wf1 = (const float*)d_in[20];
    const float* bf1 = (const float*)d_in[21];
    const float* Wf2 = (const float*)d_in[22];
    const float* bf2 = (const float*)d_in[23];

    // workspace layout
    char* ws = (char*)d_ws;
    size_t off = 0;
    float* dinv = (float*)(ws + off); off += (size_t)N * sizeof(float);
    off = (off + 255) & ~(size_t)255;
    float* bufA = (float*)(ws + off); off += (size_t)N * DFEAT * sizeof(float);
    float* bufB = (float*)(ws + off); off += (size_t)N * DFEAT * sizeof(float);
    float* bufX = (float*)(ws + off);

    const int T = 256;
    constexpr int RT = 4;
    const int rowBlocks = (N + 16 * RT - 1) / (16 * RT);

    // normalization coefficients
    deg_init_kernel<<<(N + T - 1) / T, T, 0, stream>>>(dinv, N);
    deg_count_kernel<<<(E + T - 1) / T, T, 0, stream>>>(edst, dinv, E);
    deg_finalize_kernel<<<(N + T - 1) / T, T, 0, stream>>>(dinv, N);

    const long long nThreadsE = (long long)E * 32;
    const int gE = (int)((nThreadsE + T - 1) / T);

    const float* xin = x;
    for (int l = 0; l < 3; ++l) {
        // h = x @ Wc ; fused epilogue initializes bufB = dinv^2*h + bc (self loop)
        gemm_wmma_kernel<128, RT><<<rowBlocks, 256, 0, stream>>>(
            xin, Wc[l], nullptr, bufA, N, 0, dinv, bc[l], bufB);
        // edge scatter-add into bufB
        agg_edges_kernel<<<gE, T, 0, stream>>>(bufA, dinv, esrc, edst, bufB, E);
        // MLP: relu(h@Wa+ba) @ Wb + bb, then outer relu -> next x
        gemm_wmma_kernel<128, RT><<<rowBlocks, 256, 0, stream>>>(
            bufB, Wa[l], ba[l], bufA, N, 1, nullptr, nullptr, nullptr);
        gemm_wmma_kernel<128, RT><<<rowBlocks, 256, 0, stream>>>(
            bufA, Wb[l], bb[l], bufX, N, 1, nullptr, nullptr, nullptr);
        xin = bufX;
    }

    // final: relu(x @ Wf1 + bf1) -> [N,64], then sigmoid(dot + bf2)
    gemm_wmma_kernel<64, RT><<<rowBlocks, 128, 0, stream>>>(
        xin, Wf1, bf1, bufA, N, 1, nullptr, nullptr, nullptr);
    head_kernel<<<(N + T - 1) / T, T, 0, stream>>>(bufA, Wf2, bf2, (float*)d_out, N);
}